// FeatAug_34514357190764
// MI455X (gfx1250) — compile-verified
//
#include <hip/hip_runtime.h>

typedef float v4f __attribute__((ext_vector_type(4)));

#define N_ROWS          16384
#define D_COLS          2048
#define N_PAIRS         (N_ROWS / 2)                 // 8192
#define PAIRS_PER_BLOCK 4
#define GRID_BLOCKS     (N_PAIRS / PAIRS_PER_BLOCK)  // 2048
#define BLOCK_THREADS   256
#define STAGE_FLOATS    (4 * D_COLS)                 // 4 rows per stage = 32 KB
#define STAGE_BYTES     (STAGE_FLOATS * 4)
#define AUG_SEL         0                            // stand-in for jax.random aug index

// --- cheap stateless hash; one hash yields 4 mask bits (one float4 chunk) ---
__device__ __forceinline__ unsigned h32(unsigned x) {
    x ^= x >> 16; x *= 0x7feb352dU;
    x ^= x >> 15; x *= 0x846ca68bU;
    x ^= x >> 16;
    return x;
}

// low 32 bits of a generic pointer into LDS == byte offset within the
// workgroup's LDS allocation (flat->LDS aperture mapping truncates high bits)
__device__ __forceinline__ unsigned lds_off(const void* p) {
    return (unsigned)(unsigned long long)p;
}

// CDNA5 async copy engine, GVS addressing: one SGPR64 base + one shared VGPR32
// offset. INST_OFFSET is added to BOTH the LDS address and the memory address
// (CDNA5 ISA), and our LDS layout mirrors the 2-row memory
// layout (row pitch 8192 B, chunk pitch 4096 B), so offsets {0,4096,8192,12288}
// move an entire row pair with zero per-lane 64-bit address math.
__device__ __forceinline__ void async_pair(unsigned lds_base, unsigned voff,
                                           const float* row0) {
    asm volatile(
        "global_load_async_to_lds_b128 %0, %1, %2 offset:0\n\t"
        "global_load_async_to_lds_b128 %0, %1, %2 offset:4096\n\t"
        "global_load_async_to_lds_b128 %0, %1, %2 offset:8192\n\t"
        "global_load_async_to_lds_b128 %0, %1, %2 offset:12288"
        :
        : "v"(lds_base), "v"(voff), "s"((unsigned long long)row0)
        : "memory");
}

__global__ void __launch_bounds__(BLOCK_THREADS)
feataug_kernel(const float* __restrict__ feats,
               const float* __restrict__ aug,        // selected aug slice, N x D
               const long long* __restrict__ targets,
               float* __restrict__ out_inter,        // normalized inter_feat
               float* __restrict__ out_inner)        // normalized inner_feat
{
    __shared__ float smem[2 * STAGE_FLOATS];   // 64 KB double buffer
    __shared__ float red[2][4];                // double-buffered norm accumulators

    const unsigned t    = threadIdx.x;
    const unsigned bid  = blockIdx.x;
    const unsigned lane = t & 31u;

    const unsigned voff  = t * 16u;                  // byte offset of this lane's float4
    const unsigned lds0  = lds_off(&smem[0]) + voff; // stage0 feats-pair dst
    // stage layout: [feats r0 | feats r1 | aug r0 | aug r1], 8 KB each

    if (t < 8) ((float*)red)[t] = 0.f;

    // prime stage 0
    {
        const unsigned r0 = 2u * (bid * PAIRS_PER_BLOCK);
        async_pair(lds0,              voff, feats + (size_t)r0 * D_COLS);
        async_pair(lds0 + 16384u,     voff, aug   + (size_t)r0 * D_COLS);
    }

    unsigned cur = 0;
#pragma unroll
    for (int i = 0; i < PAIRS_PER_BLOCK; ++i) {
        const unsigned p   = bid * PAIRS_PER_BLOCK + (unsigned)i;
        const unsigned nxt = cur ^ 1u;

        if (i + 1 < PAIRS_PER_BLOCK) {
            const unsigned r0n  = 2u * (p + 1u);
            const unsigned ldsn = lds0 + nxt * (unsigned)STAGE_BYTES;
            async_pair(ldsn,          voff, feats + (size_t)r0n * D_COLS);
            async_pair(ldsn + 16384u, voff, aug   + (size_t)r0n * D_COLS);
            asm volatile("s_wait_asynccnt 0x8" ::: "memory");  // current stage landed
        } else {
            asm volatile("s_wait_asynccnt 0x0" ::: "memory");
        }
        __syncthreads();                       // B1: tile + red[cur] reset visible

        if (t < 4) red[nxt][t] = 0.f;          // prep accumulators for next iter

        const unsigned r0 = 2u * p, r1 = r0 + 1u;
        const bool same = (targets[r0] == targets[r1]);

        const float* sb = &smem[cur * STAGE_FLOATS];
        v4f vi0[2], vi1[2], vA[2], vB[2];
        float ssi0 = 0.f, ssi1 = 0.f, ssA = 0.f, ssB = 0.f;

#pragma unroll
        for (int c = 0; c < 2; ++c) {
            const unsigned col  = (unsigned)c * 1024u + t * 4u;
            const unsigned col4 = (unsigned)c * 256u + t;       // float4 index in row
            v4f f0 = *(const v4f*)&sb[0 * D_COLS + col];
            v4f f1 = *(const v4f*)&sb[1 * D_COLS + col];
            v4f g0 = *(const v4f*)&sb[2 * D_COLS + col];
            v4f g1 = *(const v4f*)&sb[3 * D_COLS + col];

            const unsigned hm0 = h32(0x68bc21ebu ^ (r0 * 512u + col4));
            const unsigned hm1 = h32(0x68bc21ebu ^ (r1 * 512u + col4));
            const unsigned hmp = h32(0x1b873593u ^ (p  * 512u + col4));

            v4f i0, i1, A, B;
#pragma unroll
            for (int e = 0; e < 4; ++e) {
                const bool m0 = (hm0 >> e) & 1u;
                const bool m1 = (hm1 >> e) & 1u;
                const bool mp = (hmp >> e) & 1u;
                i0[e] = m0 ? f0[e] : g0[e];
                i1[e] = m1 ? f1[e] : g1[e];
                const float a = mp ? f0[e] : f1[e];
                const float b = mp ? f1[e] : f0[e];
                A[e] = same ? a : 0.f;
                B[e] = same ? b : 0.f;
            }
            vi0[c] = i0; vi1[c] = i1; vA[c] = A; vB[c] = B;
            ssi0 += i0.x*i0.x + i0.y*i0.y + i0.z*i0.z + i0.w*i0.w;
            ssi1 += i1.x*i1.x + i1.y*i1.y + i1.z*i1.z + i1.w*i1.w;
            ssA  += A.x*A.x + A.y*A.y + A.z*A.z + A.w*A.w;
            ssB  += B.x*B.x + B.y*B.y + B.z*B.z + B.w*B.w;
        }

        // wave32 butterfly reduction
#pragma unroll
        for (int off = 16; off > 0; off >>= 1) {
            ssi0 += __shfl_xor(ssi0, off, 32);
            ssi1 += __shfl_xor(ssi1, off, 32);
            ssA  += __shfl_xor(ssA , off, 32);
            ssB  += __shfl_xor(ssB , off, 32);
        }
        if (lane == 0) {
            atomicAdd(&red[cur][0], ssi0);
            atomicAdd(&red[cur][1], ssi1);
            atomicAdd(&red[cur][2], ssA);
            atomicAdd(&red[cur][3], ssB);
        }
        __syncthreads();                       // B2: block-wide sums complete

        const float inv_i0 = 1.f / fmaxf(sqrtf(red[cur][0]), 1e-12f);
        const float inv_i1 = 1.f / fmaxf(sqrtf(red[cur][1]), 1e-12f);
        const float inv_A  = 1.f / fmaxf(sqrtf(red[cur][2]), 1e-12f);
        const float inv_B  = 1.f / fmaxf(sqrtf(red[cur][3]), 1e-12f);

#pragma unroll
        for (int c = 0; c < 2; ++c) {
            const unsigned col = (unsigned)c * 1024u + t * 4u;
            __builtin_nontemporal_store(vA[c]  * inv_A,
                (v4f*)&out_inter[(size_t)r0 * D_COLS + col]);
            __builtin_nontemporal_store(vB[c]  * inv_B,
                (v4f*)&out_inter[(size_t)r1 * D_COLS + col]);
            __builtin_nontemporal_store(vi0[c] * inv_i0,
                (v4f*)&out_inner[(size_t)r0 * D_COLS + col]);
            __builtin_nontemporal_store(vi1[c] * inv_i1,
                (v4f*)&out_inner[(size_t)r1 * D_COLS + col]);
        }

        cur = nxt;
    }
}

extern "C" void kernel_launch(void* const* d_in, const int* in_sizes, int n_in,
                              void* d_out, int out_size, void* d_ws, size_t ws_size,
                              hipStream_t stream) {
    (void)in_sizes; (void)n_in; (void)out_size; (void)d_ws; (void)ws_size;
    const float*     feats = (const float*)d_in[0];
    const float*     augs  = (const float*)d_in[1];
    const long long* tgts  = (const long long*)d_in[2];
    const float*     aug   = augs + (size_t)AUG_SEL * N_ROWS * D_COLS;

    float* out_inter = (float*)d_out;
    float* out_inner = out_inter + (size_t)N_ROWS * D_COLS;

    feataug_kernel<<<GRID_BLOCKS, BLOCK_THREADS, 0, stream>>>(
        feats, aug, tgts, out_inter, out_inner);
}